// MoEBiEncoder_41266045780089
// MI455X (gfx1250) — compile-verified
//
#include <hip/hip_runtime.h>
#include <math.h>

typedef __attribute__((ext_vector_type(2))) float v2f;
typedef __attribute__((ext_vector_type(8))) float v8f;

#define H   768
#define H2  384
#define E   8
#define TILE_M 32
#define XPITCH 772   // 768 + 4 pad -> bank-conflict-free column reads
#define HPITCH 388   // 384 + 4 pad

// ---------------- kernel 1: zero the per-expert counters ----------------
__global__ void init_counts(int* __restrict__ cnt) {
  if (threadIdx.x < E) cnt[threadIdx.x] = 0;
}

// ---------------- kernel 2: softmax + top-1 routing ----------------
__global__ void gate_kernel(const float* __restrict__ logits,
                            float* __restrict__ gates,
                            int* __restrict__ cnt,
                            int* __restrict__ bucket, int B) {
  int b = blockIdx.x * blockDim.x + threadIdx.x;
  if (b >= B) return;
  const float4* lp = (const float4*)(logits + (size_t)b * E);
  float4 p0 = lp[0], p1 = lp[1];
  float l[E] = {p0.x, p0.y, p0.z, p0.w, p1.x, p1.y, p1.z, p1.w};
  float m = l[0]; int ei = 0;
#pragma unroll
  for (int i = 1; i < E; ++i) if (l[i] > m) { m = l[i]; ei = i; }  // first-max, matches argmax
  float s = 0.f;
#pragma unroll
  for (int i = 0; i < E; ++i) s += expf(l[i] - m);
  gates[b] = 1.0f / s;                      // softmax prob of the argmax expert
  int pos = atomicAdd(&cnt[ei], 1);
  bucket[ei * B + pos] = b;
}

// ---------------- kernel 3: grouped-GEMM MLP with f32 WMMA ----------------
// Block: 256 threads (8 wave32s), M-tile = 32 rows of one expert's bucket.
// Each wave owns a column range and BOTH 16-row M-tiles, so every loaded
// weight (B) fragment feeds two v_wmma_f32_16x16x4_f32 ops.
__global__ __launch_bounds__(256)
void moe_mlp_kernel(const float* __restrict__ emb,
                    const float* __restrict__ W1, const float* __restrict__ b1,
                    const float* __restrict__ W2, const float* __restrict__ b2,
                    const float* __restrict__ gates,
                    const int* __restrict__ cnt,
                    const int* __restrict__ bucket,
                    float* __restrict__ out, int B) {
  const int e = blockIdx.y;
  const int nrows = cnt[e];
  const int r0 = blockIdx.x * TILE_M;
  if (r0 >= nrows) return;
  const int valid = min(TILE_M, nrows - r0);

  extern __shared__ float smem[];
  float* Xs = smem;                               // [32][XPITCH]
  float* Hs = smem + TILE_M * XPITCH;             // [32][HPITCH]
  float* Ys = Hs + TILE_M * HPITCH;               // [32][XPITCH]
  __shared__ int   sidx[TILE_M];
  __shared__ float sg[TILE_M];

  const int tid = threadIdx.x;
  if (tid < TILE_M) {
    int r = (tid < valid) ? (r0 + tid) : r0;      // pad with row 0 (safe addresses)
    int sidx_t = bucket[e * B + r];
    sidx[tid] = sidx_t;
    sg[tid] = gates[sidx_t];
  }
  __syncthreads();

  // gather 32 embedding rows into LDS (float4, coalesced)
  for (int i = tid; i < TILE_M * (H / 4); i += 256) {
    int row = i / (H / 4);
    int c4  = i - row * (H / 4);
    float4 v = ((const float4*)(emb + (size_t)sidx[row] * H))[c4];
    float* dst = &Xs[row * XPITCH + c4 * 4];
    dst[0] = v.x; dst[1] = v.y; dst[2] = v.z; dst[3] = v.w;
  }
  __syncthreads();

  const int lane = tid & 31;
  const int wave = tid >> 5;
  const int mn = lane & 15;      // A row / B,D column
  const int hf = lane >> 4;      // lane half: selects K pair / D row half

  const float* W1e = W1 + (size_t)e * H * H2;
  const float* b1e = b1 + (size_t)e * H2;
  const float* W2e = W2 + (size_t)e * H2 * H;
  const float* b2e = b2 + (size_t)e * H;

  // ---- layer 1: Hs = relu(X @ W1e + b1e), [32 x 384]
  // wave -> 3 N-tiles (48 cols) x 2 M-tiles; weights read exactly once/block
  {
    v8f acc[2][3];
    v8f zero = {};
#pragma unroll
    for (int m = 0; m < 2; ++m)
#pragma unroll
      for (int t = 0; t < 3; ++t) acc[m][t] = zero;
    const int n0 = wave * 48;
    for (int k = 0; k < H; k += 4) {
      v2f a0 = *(const v2f*)&Xs[mn * XPITCH + k + 2 * hf];          // rows 0..15
      v2f a1 = *(const v2f*)&Xs[(16 + mn) * XPITCH + k + 2 * hf];   // rows 16..31
      const float* wp = W1e + (size_t)(k + 2 * hf) * H2 + n0 + mn;
#pragma unroll
      for (int t = 0; t < 3; ++t) {
        v2f bv;
        bv.x = wp[t * 16];
        bv.y = wp[H2 + t * 16];
        acc[0][t] = __builtin_amdgcn_wmma_f32_16x16x4_f32(false, a0, false, bv, (short)0, acc[0][t], false, false);
        acc[1][t] = __builtin_amdgcn_wmma_f32_16x16x4_f32(false, a1, false, bv, (short)0, acc[1][t], false, false);
      }
    }
#pragma unroll
    for (int t = 0; t < 3; ++t) {
      int col = n0 + t * 16 + mn;
      float bb = b1e[col];
#pragma unroll
      for (int j = 0; j < 8; ++j) {
        int row = j + 8 * hf;                                       // D layout: M = j + 8*half
        Hs[row * HPITCH + col]        = fmaxf(acc[0][t][j] + bb, 0.f);
        Hs[(16 + row) * HPITCH + col] = fmaxf(acc[1][t][j] + bb, 0.f);
      }
    }
  }
  __syncthreads();

  // ---- layer 2: Ys = (H @ W2e + b2e) * gate, [32 x 768]
  // wave -> 6 N-tiles (96 cols) x 2 M-tiles; weights read exactly once/block
  {
    v8f acc[2][6];
    v8f zero = {};
#pragma unroll
    for (int m = 0; m < 2; ++m)
#pragma unroll
      for (int t = 0; t < 6; ++t) acc[m][t] = zero;
    const int n0 = wave * 96;
    for (int k = 0; k < H2; k += 4) {
      v2f a0 = *(const v2f*)&Hs[mn * HPITCH + k + 2 * hf];
      v2f a1 = *(const v2f*)&Hs[(16 + mn) * HPITCH + k + 2 * hf];
      const float* wp = W2e + (size_t)(k + 2 * hf) * H + n0 + mn;
#pragma unroll
      for (int t = 0; t < 6; ++t) {
        v2f bv;
        bv.x = wp[t * 16];
        bv.y = wp[H + t * 16];
        acc[0][t] = __builtin_amdgcn_wmma_f32_16x16x4_f32(false, a0, false, bv, (short)0, acc[0][t], false, false);
        acc[1][t] = __builtin_amdgcn_wmma_f32_16x16x4_f32(false, a1, false, bv, (short)0, acc[1][t], false, false);
      }
    }
#pragma unroll
    for (int t = 0; t < 6; ++t) {
      int c = n0 + t * 16 + mn;
      float bb = b2e[c];
#pragma unroll
      for (int j = 0; j < 8; ++j) {
        int row = j + 8 * hf;
        Ys[row * XPITCH + c]        = (acc[0][t][j] + bb) * sg[row];
        Ys[(16 + row) * XPITCH + c] = (acc[1][t][j] + bb) * sg[16 + row];
      }
    }
  }
  __syncthreads();

  // ---- per-row: l2norm(eps=1e-6) -> +X -> l2norm(eps=1e-12) -> scatter
#pragma unroll
  for (int rr = 0; rr < 4; ++rr) {
    int row = wave * 4 + rr;                    // 8 waves x 4 rows = 32
    float s = 0.f;
    for (int c = lane; c < H; c += 32) { float y = Ys[row * XPITCH + c]; s += y * y; }
#pragma unroll
    for (int off = 16; off > 0; off >>= 1) s += __shfl_xor(s, off, 32);
    float inv1 = 1.0f / fmaxf(sqrtf(s), 1e-6f);
    float s2 = 0.f;
    for (int c = lane; c < H; c += 32) {
      float v = Ys[row * XPITCH + c] * inv1 + Xs[row * XPITCH + c];
      Ys[row * XPITCH + c] = v;
      s2 += v * v;
    }
#pragma unroll
    for (int off = 16; off > 0; off >>= 1) s2 += __shfl_xor(s2, off, 32);
    float inv2 = 1.0f / fmaxf(sqrtf(s2), 1e-12f);
    if (row < valid) {
      float* op = out + (size_t)sidx[row] * H;
      for (int c = lane; c < H; c += 32) op[c] = Ys[row * XPITCH + c] * inv2;
    }
  }
}

extern "C" void kernel_launch(void* const* d_in, const int* in_sizes, int n_in,
                              void* d_out, int out_size, void* d_ws, size_t ws_size,
                              hipStream_t stream) {
  const float* emb    = (const float*)d_in[0];
  const float* logits = (const float*)d_in[1];
  const float* W1     = (const float*)d_in[2];
  const float* b1     = (const float*)d_in[3];
  const float* W2     = (const float*)d_in[4];
  const float* b2     = (const float*)d_in[5];
  float* out = (float*)d_out;
  const int B = in_sizes[1] / E;

  // workspace layout: cnt[E] @0 | gates[B] @256 | bucket[E*B] after
  char* ws = (char*)d_ws;
  int*   cnt    = (int*)ws;
  float* gates  = (float*)(ws + 256);
  int*   bucket = (int*)(ws + 256 + (size_t)B * sizeof(float));

  hipLaunchKernelGGL(init_counts, dim3(1), dim3(32), 0, stream, cnt);
  hipLaunchKernelGGL(gate_kernel, dim3((B + 255) / 256), dim3(256), 0, stream,
                     logits, gates, cnt, bucket, B);

  const int tiles = (B + TILE_M - 1) / TILE_M;
  const size_t smem = (size_t)(TILE_M * XPITCH + TILE_M * HPITCH + TILE_M * XPITCH) * sizeof(float);
  hipLaunchKernelGGL(moe_mlp_kernel, dim3(tiles, E), dim3(256), smem, stream,
                     emb, W1, b1, W2, b2, gates, cnt, bucket, out, B);
}